// CausalGraphNetwork_50259707297820
// MI455X (gfx1250) — compile-verified
//
#include <hip/hip_runtime.h>
#include <hip/hip_bf16.h>

// ---------------------------------------------------------------------------
// CausalGraphNetwork on gfx1250 (MI455X)
//  - fp32 WMMA (V_WMMA_F32_16X16X4_F32) for all matmuls
//  - Tensor Data Mover (tensor_load_to_lds + s_wait_tensorcnt) for tile staging
// B=4, S=256, V=64, L=10 (LP1=11), H=128, NH=8, DH=16, NL=3
// ---------------------------------------------------------------------------

#define CB 4
#define CS 256
#define CV 64
#define CLP1 11
#define CH 128
#define CNH 8
#define CDH 16
#define CNL 3
#define CBT (CB * CS)          // 1024
#define CM  (CBT * CV)         // 65536 token-variable rows
#define CKXC 80                // 64 (src) + 11 (lag) padded to 80

typedef float v2f_t __attribute__((ext_vector_type(2)));
typedef float v8f_t __attribute__((ext_vector_type(8)));
typedef unsigned int u32x4_t __attribute__((ext_vector_type(4)));
typedef int i32x4_t __attribute__((ext_vector_type(4)));
typedef int i32x8_t __attribute__((ext_vector_type(8)));

__device__ inline v8f_t wmma4(v2f_t a, v2f_t b, v8f_t c) {
  // D(16x16,f32) = A(16x4,f32) * B(4x16,f32) + C
  return __builtin_amdgcn_wmma_f32_16x16x4_f32(false, a, false, b, (short)0, c,
                                               false, false);
}

// ---------------------------------------------------------------------------
// TDM: DMA a 2-D tile (nrows x row_elems f32, global row stride in elements)
// into LDS at byte offset lds_off, rows packed contiguously.
// D# layout per CDNA5 ISA ch.8 (group0/group1; groups 2-3 zero for 2-D).
// ---------------------------------------------------------------------------
__device__ inline void tdm_load_2d(unsigned lds_off, const void* gaddr,
                                   int row_elems, int nrows,
                                   int row_stride_elems) {
  unsigned long long ga = (unsigned long long)gaddr;
  u32x4_t g0;
  g0.x = 1u;                                  // count=1 (valid user descriptor)
  g0.y = lds_off;                             // lds_addr (bytes)
  g0.z = (unsigned)ga;                        // global_addr[31:0]
  g0.w = (unsigned)(ga >> 32) | (2u << 30);   // global_addr[56:32] | type=2
  i32x8_t g1;
  g1[0] = 0x20000;                            // data_size=2 (4B); mask/flags=0
  g1[1] = row_elems << 16;                    // tensor_dim0[15:0] @ bits[63:48]
  g1[2] = nrows << 16;                        // tensor_dim1[15:0] @ bits[95:80]
  g1[3] = row_elems << 16;                    // tile_dim0 @ bits[127:112]
  g1[4] = nrows;                              // tile_dim1 (tile_dim2 = 0)
  g1[5] = row_stride_elems;                   // tensor_dim0_stride[31:0]
  g1[6] = 0;                                  // stride hi / dim1_stride lo
  g1[7] = 0;                                  // dim1_stride hi
  i32x4_t gz = {0, 0, 0, 0};
#if __clang_major__ >= 23
  i32x8_t gz8 = {0, 0, 0, 0, 0, 0, 0, 0};
  __builtin_amdgcn_tensor_load_to_lds(g0, g1, gz, gz, gz8, 0);
#else
  __builtin_amdgcn_tensor_load_to_lds(g0, g1, gz, gz, 0);
#endif
}

__device__ inline unsigned lds_offset_of(const void* p) {
  // gfx1250 flat-LDS aperture keeps the LDS byte offset in addr[31:0]
  return (unsigned)(unsigned long long)p;
}

// ---------------------------------------------------------------------------
// Elementwise: adj = sigmoid(adjacency_logits)
// ---------------------------------------------------------------------------
__global__ void sigmoid_kernel(const float* __restrict__ in,
                               float* __restrict__ out, int n) {
  int i = blockIdx.x * blockDim.x + threadIdx.x;
  if (i < n) out[i] = 1.0f / (1.0f + expf(-in[i]));
}

// ---------------------------------------------------------------------------
// E (80x128): rows 0..63 = var_emb, rows 64..74 = temp_emb, 75..79 = zero
// ---------------------------------------------------------------------------
__global__ void build_E_kernel(const float* __restrict__ ve,
                               const float* __restrict__ te,
                               float* __restrict__ E) {
  int i = blockIdx.x * blockDim.x + threadIdx.x;
  if (i >= CKXC * CH) return;
  int r = i >> 7, c = i & 127;
  float val = 0.0f;
  if (r < CV) val = ve[r * CH + c];
  else if (r < CV + CLP1) val = te[(r - CV) * CH + c];
  E[i] = val;
}

// ---------------------------------------------------------------------------
// Xc (M x 80): per (b,t):
//   cols 0..63   : A[i,s]  = sum_l xlag[s,l] * adj[s,i,l]
//   cols 64..74  : Bl[i,l] = sum_s xlag[s,l] * adj[s,i,l]
//   cols 75..79  : zero
// row index m = bt*V + i  (matches (b,s,v,·) activation layout)
// ---------------------------------------------------------------------------
__global__ __launch_bounds__(256) void build_xc_kernel(
    const float* __restrict__ x, const float* __restrict__ adj,
    float* __restrict__ Xc) {
  __shared__ float xl[CV][CLP1 + 1];
  int bt = blockIdx.x;
  int b = bt / CS, t = bt % CS;
  int tid = threadIdx.x;

  for (int idx = tid; idx < CV * CLP1; idx += 256) {
    int s = idx / CLP1, l = idx % CLP1;
    xl[s][l] = (t - l >= 0) ? x[((long)b * CS + (t - l)) * CV + s] : 0.0f;
  }
  __syncthreads();

  for (int idx = tid; idx < CV * CV; idx += 256) {
    int i = idx >> 6, s = idx & 63;
    const float* ap = adj + ((long)s * CV + i) * CLP1;
    float acc = 0.0f;
#pragma unroll
    for (int l = 0; l < CLP1; l++) acc += xl[s][l] * ap[l];
    Xc[((long)bt * CV + i) * CKXC + s] = acc;
  }
  for (int idx = tid; idx < CV * 16; idx += 256) {
    int i = idx >> 4, c = idx & 15;
    float acc = 0.0f;
    if (c < CLP1) {
      for (int s = 0; s < CV; s++)
        acc += xl[s][c] * adj[((long)s * CV + i) * CLP1 + c];
    }
    Xc[((long)bt * CV + i) * CKXC + CV + c] = acc;
  }
}

// ---------------------------------------------------------------------------
// Unified WMMA GEMM, N fixed at 128. Block = 256 threads = 8 waves; wave w
// produces the 16x16 output tile at columns [16w,16w+16). The shared 16xK
// A-tile is DMA'd into LDS once by the TDM (wave 0), then all 8 waves read
// A-fragments from LDS (removes the 8x-redundant global loads).
//   mode 0: out = A@W
//   mode 1: out = GELU(LayerNorm(A@W + bias))   (mechanism layers)
//   mode 2: out = A@W + bias                    (q/k/v/o projections)
// ---------------------------------------------------------------------------
__global__ __launch_bounds__(256) void gemm_n128_kernel(
    const float* __restrict__ A, long rowsA, long varA,
    const float* __restrict__ W, long varW,
    const float* __restrict__ bias, long varBias,
    const float* __restrict__ lng, const float* __restrict__ lnb, long varLN,
    float* __restrict__ out, long rowsO, long varO, int K, int mode) {
  __shared__ float Alds[16 * CH];  // 16 x K tile, rows packed (K <= 128)
  int var = blockIdx.x;
  int m0 = blockIdx.y * 16;
  int tid = threadIdx.x;
  int wv = tid >> 5, lane = tid & 31;
  int half = lane >> 4, ln16 = lane & 15;
  int n0 = wv * 16;

  if (wv == 0) {
    tdm_load_2d(lds_offset_of(Alds),
                A + (long)var * varA + (long)m0 * rowsA, K, 16, (int)rowsA);
    __builtin_amdgcn_s_wait_tensorcnt(0);
  }
  __syncthreads();

  const float* Ap = Alds + ln16 * K + half * 2;
  const float* Wp = W + (long)var * varW + (long)(half * 2) * CH + n0 + ln16;

  v8f_t acc = {0.f, 0.f, 0.f, 0.f, 0.f, 0.f, 0.f, 0.f};
  for (int k = 0; k < K; k += 4) {
    v2f_t a = *(const v2f_t*)(Ap + k);       // A[m, k+2h], A[m, k+2h+1] (LDS)
    v2f_t b;
    b.x = Wp[(long)k * CH];                  // W[k+2h,   n]
    b.y = Wp[(long)k * CH + CH];             // W[k+2h+1, n]
    acc = wmma4(a, b, acc);
  }

  float bval = 0.0f;
  if (mode != 0) bval = bias[(long)var * varBias + n0 + ln16];

  if (mode == 1) {
    __shared__ float tile[16][CH + 4];
#pragma unroll
    for (int r = 0; r < 8; r++) tile[r + half * 8][n0 + ln16] = acc[r] + bval;
    __syncthreads();
    const float* gp = lng + (long)var * varLN;
    const float* bp = lnb + (long)var * varLN;
#pragma unroll
    for (int rr = 0; rr < 2; rr++) {
      int row = wv * 2 + rr;
      float s = 0.0f, s2 = 0.0f;
      for (int c = lane; c < CH; c += 32) {
        float xv = tile[row][c];
        s += xv;
        s2 += xv * xv;
      }
#pragma unroll
      for (int off = 16; off > 0; off >>= 1) {
        s += __shfl_xor(s, off, 32);
        s2 += __shfl_xor(s2, off, 32);
      }
      float mu = s * (1.0f / CH);
      float vr = s2 * (1.0f / CH) - mu * mu;
      float rstd = rsqrtf(vr + 1e-5f);
      float* op = out + (long)var * varO + (long)(m0 + row) * rowsO;
      for (int c = lane; c < CH; c += 32) {
        float xv = (tile[row][c] - mu) * rstd * gp[c] + bp[c];
        xv = 0.5f * xv * (1.0f + erff(xv * 0.70710678118654752f));
        op[c] = xv;
      }
    }
  } else {
    float* op = out + (long)var * varO + n0 + ln16;
#pragma unroll
    for (int r = 0; r < 8; r++)
      op[(long)(m0 + r + half * 8) * rowsO] = acc[r] + bval;
  }
}

// ---------------------------------------------------------------------------
// Attention: one wave per (16 query rows, head, b*v). K and V tiles (256x16)
// are DMA'd into LDS by the TDM; scores (16x256) live in registers; softmax
// via intra-half shfl_xor; probabilities staged through LDS for P·V.
// ---------------------------------------------------------------------------
__global__ __launch_bounds__(32) void attn_kernel(
    const float* __restrict__ q, const float* __restrict__ k,
    const float* __restrict__ v, float* __restrict__ o) {
  __shared__ float at[16 * CS];       // probabilities (16 x 256)
  __shared__ float Klds[CS * CDH];    // K tile (256 x 16)
  __shared__ float Vlds[CS * CDH];    // V tile (256 x 16)
  int lane = threadIdx.x & 31;
  int half = lane >> 4, ln16 = lane & 15;
  int q0 = blockIdx.x * 16;
  int nh = blockIdx.y;
  int bv = blockIdx.z;
  int b = bv >> 6, vv = bv & 63;
  const long rs = (long)CV * CH;  // row stride between sequence positions
  const long base = ((long)b * CS * CV + vv) * CH + nh * CDH;

  // DMA K and V head-slices into LDS (single wave: wait, no barrier needed)
  tdm_load_2d(lds_offset_of(Klds), k + base, CDH, CS, (int)rs);
  tdm_load_2d(lds_offset_of(Vlds), v + base, CDH, CS, (int)rs);

  // Q fragments for this wave's 16 rows (K = 16 -> 4 fragments)
  v2f_t qa[4];
#pragma unroll
  for (int ks = 0; ks < 4; ks++)
    qa[ks] =
        *(const v2f_t*)(q + base + (long)(q0 + ln16) * rs + ks * 4 + half * 2);

  __builtin_amdgcn_s_wait_tensorcnt(0);

  // scores = Q K^T  (16 x 256)
  v8f_t sc[16];
#pragma unroll
  for (int nt = 0; nt < 16; nt++) {
    v8f_t acc = {0.f, 0.f, 0.f, 0.f, 0.f, 0.f, 0.f, 0.f};
    const float* kp = Klds + (nt * 16 + ln16) * CDH + half * 2;
#pragma unroll
    for (int ks = 0; ks < 4; ks++) {
      v2f_t bb = *(const v2f_t*)(kp + ks * 4);
      acc = wmma4(qa[ks], bb, acc);
    }
    sc[nt] = acc;
  }

  // softmax over keys (scale 1/sqrt(16) = 0.25)
  const float scl = 0.25f;
#pragma unroll
  for (int r = 0; r < 8; r++) {
    float mx = -3.4e38f;
#pragma unroll
    for (int nt = 0; nt < 16; nt++) {
      sc[nt][r] *= scl;
      mx = fmaxf(mx, sc[nt][r]);
    }
#pragma unroll
    for (int off = 8; off > 0; off >>= 1)
      mx = fmaxf(mx, __shfl_xor(mx, off, 32));
    float sm = 0.0f;
#pragma unroll
    for (int nt = 0; nt < 16; nt++) {
      float e = expf(sc[nt][r] - mx);
      sc[nt][r] = e;
      sm += e;
    }
#pragma unroll
    for (int off = 8; off > 0; off >>= 1) sm += __shfl_xor(sm, off, 32);
    float inv = 1.0f / sm;
    int row = r + half * 8;
#pragma unroll
    for (int nt = 0; nt < 16; nt++)
      at[row * CS + nt * 16 + ln16] = sc[nt][r] * inv;
  }
  __syncthreads();

  // O = P V  (16 x 16, K = 256)
  v8f_t oc = {0.f, 0.f, 0.f, 0.f, 0.f, 0.f, 0.f, 0.f};
  const float* ap2 = at + ln16 * CS + half * 2;
  for (int kk = 0; kk < CS; kk += 4) {
    v2f_t aa = *(const v2f_t*)(ap2 + kk);
    v2f_t bb;
    bb.x = Vlds[kk * CDH + ln16];
    bb.y = Vlds[(kk + 1) * CDH + ln16];
    oc = wmma4(aa, bb, oc);
  }
#pragma unroll
  for (int r = 0; r < 8; r++)
    o[base + (long)(q0 + r + half * 8) * rs + ln16] = oc[r];
}

// ---------------------------------------------------------------------------
// Output head: pred[row] = dot(o[row,:], out_W[v,:]) + out_b[v], row = bt*V+v
// ---------------------------------------------------------------------------
__global__ __launch_bounds__(256) void head_kernel(
    const float* __restrict__ o, const float* __restrict__ outW,
    const float* __restrict__ outb, float* __restrict__ pred) {
  int row = blockIdx.x * 256 + threadIdx.x;
  if (row >= CM) return;
  int vv = row & 63;
  const float* op = o + (long)row * CH;
  const float* wp = outW + vv * CH;
  float acc = outb[vv];
  for (int h = 0; h < CH; h++) acc += op[h] * wp[h];
  pred[row] = acc;
}

// ---------------------------------------------------------------------------
extern "C" void kernel_launch(void* const* d_in, const int* in_sizes, int n_in,
                              void* d_out, int out_size, void* d_ws,
                              size_t ws_size, hipStream_t stream) {
  const float* x       = (const float*)d_in[0];
  const float* adjlog  = (const float*)d_in[1];
  const float* var_emb = (const float*)d_in[2];
  const float* temp_emb= (const float*)d_in[3];
  const float* mech_W  = (const float*)d_in[4];
  const float* mech_b  = (const float*)d_in[5];
  const float* ln_g    = (const float*)d_in[6];
  const float* ln_b    = (const float*)d_in[7];
  const float* Wq      = (const float*)d_in[8];
  const float* Wk      = (const float*)d_in[9];
  const float* Wv      = (const float*)d_in[10];
  const float* Wo      = (const float*)d_in[11];
  const float* bq      = (const float*)d_in[12];
  const float* bk      = (const float*)d_in[13];
  const float* bv      = (const float*)d_in[14];
  const float* bo      = (const float*)d_in[15];
  const float* out_W   = (const float*)d_in[16];
  const float* out_b   = (const float*)d_in[17];
  float* pred = (float*)d_out;

  // workspace layout (elements, 32-aligned)
  const size_t nAdj = (size_t)CV * CV * CLP1;        // 45056
  const size_t nE   = (size_t)CKXC * CH;             // 10240
  const size_t nXc  = (size_t)CM * CKXC;             // 5,242,880
  const size_t nBuf = (size_t)CM * CH;               // 8,388,608
  const size_t need = (nAdj + nE + nXc + 5 * nBuf) * sizeof(float);
  if (ws_size < need) return;

  float* ws   = (float*)d_ws;
  float* adjs = ws;                 size_t off = nAdj;
  float* E    = ws + off;           off += nE;
  float* Xc   = ws + off;           off += nXc;
  float* bufA = ws + off;           off += nBuf;
  float* bufB = ws + off;           off += nBuf;
  float* bufQ = ws + off;           off += nBuf;
  float* bufK = ws + off;           off += nBuf;
  float* bufV = ws + off;           off += nBuf;

  const long HH = (long)CH * CH;    // 16384
  const long rowS = (long)CV * CH;  // 8192

  sigmoid_kernel<<<(int)((nAdj + 255) / 256), 256, 0, stream>>>(adjlog, adjs,
                                                                (int)nAdj);
  build_E_kernel<<<(int)((nE + 255) / 256), 256, 0, stream>>>(var_emb, temp_emb,
                                                              E);
  build_xc_kernel<<<CBT, 256, 0, stream>>>(x, adjs, Xc);

  // z = Xc @ E   (M=65536, K=80)
  gemm_n128_kernel<<<dim3(1, CM / 16), 256, 0, stream>>>(
      Xc, CKXC, 0, E, 0, nullptr, 0, nullptr, nullptr, 0,
      bufA, CH, 0, CKXC, 0);

  // 3 mechanism layers: Linear -> LayerNorm -> GELU, per variable
  float* src = bufA;
  float* dst = bufB;
  for (int li = 0; li < CNL; li++) {
    gemm_n128_kernel<<<dim3(CV, CBT / 16), 256, 0, stream>>>(
        src, rowS, CH, mech_W + (long)li * HH, (long)CNL * HH,
        mech_b + (long)li * CH, (long)CNL * CH,
        ln_g + (long)li * CH, ln_b + (long)li * CH, (long)CNL * CH,
        dst, rowS, CH, CH, 1);
    float* tmp = src; src = dst; dst = tmp;
  }
  // final z now in `src` (== bufB after 3 layers)

  // Q/K/V projections (per variable, bias epilogue)
  gemm_n128_kernel<<<dim3(CV, CBT / 16), 256, 0, stream>>>(
      src, rowS, CH, Wq, HH, bq, CH, nullptr, nullptr, 0, bufQ, rowS, CH, CH, 2);
  gemm_n128_kernel<<<dim3(CV, CBT / 16), 256, 0, stream>>>(
      src, rowS, CH, Wk, HH, bk, CH, nullptr, nullptr, 0, bufK, rowS, CH, CH, 2);
  gemm_n128_kernel<<<dim3(CV, CBT / 16), 256, 0, stream>>>(
      src, rowS, CH, Wv, HH, bv, CH, nullptr, nullptr, 0, bufV, rowS, CH, CH, 2);

  // attention -> bufA
  attn_kernel<<<dim3(CS / 16, CNH, CB * CV), 32, 0, stream>>>(bufQ, bufK, bufV,
                                                              bufA);

  // output projection: o @ Wo + bo -> bufB  (z no longer needed)
  gemm_n128_kernel<<<dim3(CV, CBT / 16), 256, 0, stream>>>(
      bufA, rowS, CH, Wo, HH, bo, CH, nullptr, nullptr, 0, bufB, rowS, CH, CH, 2);

  // output head
  head_kernel<<<CM / 256, 256, 0, stream>>>(bufB, out_W, out_b, pred);
}